// DirichletFocalLoss_84808424227237
// MI455X (gfx1250) — compile-verified
//
#include <hip/hip_runtime.h>
#include <math.h>

#define NUM_CLASSES 80

typedef float v2f __attribute__((ext_vector_type(2)));
typedef float v8f __attribute__((ext_vector_type(8)));

// ---------------------------------------------------------------------------
// Wave32 sum-reduction built on V_WMMA_F32_16X16X4_F32 (ones-matrix reduce).
// Stage 1: A vgprs both = v, B = ones  ->  D1[m][n] = 2*(v[m] + v[m+16])
//          (row-constant over n; robust to the K-slot ordering within a lane).
// Stage 2: A = ones, B fed with D1's 8 vgprs as 4 accumulating pairs. The 16
//          (vgpr, lane-half) slots of D1 are a bijection onto its 16 rows, so
//          every output element accumulates sum over all rows = 2 * sum(v).
// ---------------------------------------------------------------------------
__device__ __forceinline__ float wave_sum_wmma(float v) {
  v2f a;    a[0] = v;    a[1] = v;
  v2f ones; ones[0] = 1.0f; ones[1] = 1.0f;
  v8f z = {};
  v8f d1 = __builtin_amdgcn_wmma_f32_16x16x4_f32(false, a, false, ones,
                                                 (short)0, z, false, false);
  v8f acc = {};
  v2f bp;
  bp[0] = d1[0]; bp[1] = d1[1];
  acc = __builtin_amdgcn_wmma_f32_16x16x4_f32(false, ones, false, bp,
                                              (short)0, acc, false, false);
  bp[0] = d1[2]; bp[1] = d1[3];
  acc = __builtin_amdgcn_wmma_f32_16x16x4_f32(false, ones, false, bp,
                                              (short)0, acc, false, false);
  bp[0] = d1[4]; bp[1] = d1[5];
  acc = __builtin_amdgcn_wmma_f32_16x16x4_f32(false, ones, false, bp,
                                              (short)0, acc, false, false);
  bp[0] = d1[6]; bp[1] = d1[7];
  acc = __builtin_amdgcn_wmma_f32_16x16x4_f32(false, ones, false, bp,
                                              (short)0, acc, false, false);
  return 0.5f * acc[0];
}

__device__ __forceinline__ float digammaf_(float x) {
  // psi(x) = psi(x+1) - 1/x; asymptotic series for x >= 6.
  float r = 0.0f;
  while (x < 6.0f) { r -= 1.0f / x; x += 1.0f; }
  float inv  = 1.0f / x;
  float inv2 = inv * inv;
  r += logf(x) - 0.5f * inv
     - inv2 * (0.0833333333f - inv2 * (0.0083333333f - inv2 * 0.0039682540f));
  return r;
}

__global__ __launch_bounds__(256)
void dfl_main(const float* __restrict__ cls,
              const float* __restrict__ reg,
              const float* __restrict__ anchors,
              const float* __restrict__ ann,
              float* __restrict__ ws,
              int A, int M) {
  extern __shared__ float sann[];
  const int b   = blockIdx.y;
  const int tid = threadIdx.x;

  for (int i = tid; i < M * 5; i += blockDim.x)
    sann[i] = ann[(size_t)b * M * 5 + i];
  __syncthreads();

  const int a = blockIdx.x * blockDim.x + tid;

  float cls_acc = 0.0f, reg_acc = 0.0f, kl_acc = 0.0f, l1_acc = 0.0f, npos = 0.0f;

  if (a < A) {
    const float4 ab = ((const float4*)anchors)[a];
    const float ax1 = ab.x, ay1 = ab.y, ax2 = ab.z, ay2 = ab.w;
    const float aw = ax2 - ax1, ah = ay2 - ay1;
    const float aarea = aw * ah;

    // IoU argmax over M annotations (first-occurrence max, like jnp.argmax)
    float best = -1e30f; int bm = 0;
    for (int m = 0; m < M; ++m) {
      const float gx1 = sann[m * 5 + 0], gy1 = sann[m * 5 + 1];
      const float gx2 = sann[m * 5 + 2], gy2 = sann[m * 5 + 3];
      const float barea = (gx2 - gx1) * (gy2 - gy1);
      float iw = fminf(ax2, gx2) - fmaxf(ax1, gx1); iw = fmaxf(iw, 0.0f);
      float ih = fminf(ay2, gy2) - fmaxf(ay1, gy1); ih = fmaxf(ih, 0.0f);
      const float inter = iw * ih;
      const float ua = fmaxf(aarea + barea - inter, 1e-8f);
      const float iou = inter / ua;
      if (iou > best) { best = iou; bm = m; }
    }
    const bool positive = best >= 0.5f;
    const bool ignore   = (!positive) && (best >= 0.4f);
    const int  label    = (int)sann[bm * 5 + 4];

    // ---- focal classification loss (skip ignore anchors entirely) ----
    if (!ignore) {
      const float4* cp = (const float4*)(cls + ((size_t)b * A + a) * NUM_CLASSES);
      if (positive) {
        npos = 1.0f;
        for (int i = 0; i < NUM_CLASSES / 4; ++i) {
          const float4 v = cp[i];
          const float pv[4] = {v.x, v.y, v.z, v.w};
#pragma unroll
          for (int k = 0; k < 4; ++k) {
            const int c = i * 4 + k;
            const float p = fminf(fmaxf(pv[k], 1e-4f), 1.0f - 1e-4f);
            if (c == label) {
              const float w = 1.0f - p;
              cls_acc += 0.25f * w * w * (-logf(p));
            } else {
              cls_acc += 0.75f * p * p * (-logf(1.0f - p));
            }
          }
        }
      } else {  // pure negative: uniform zero targets
        for (int i = 0; i < NUM_CLASSES / 4; ++i) {
          const float4 v = cp[i];
          const float pv[4] = {v.x, v.y, v.z, v.w};
#pragma unroll
          for (int k = 0; k < 4; ++k) {
            const float p = fminf(fmaxf(pv[k], 1e-4f), 1.0f - 1e-4f);
            cls_acc += 0.75f * p * p * (-logf(1.0f - p));
          }
        }
      }
    }

    // ---- regression / L1 / Dirichlet-KL: positives only ----
    if (positive) {
      const float acx = ax1 + 0.5f * aw, acy = ay1 + 0.5f * ah;
      const float gx1 = sann[bm * 5 + 0], gy1 = sann[bm * 5 + 1];
      const float gx2 = sann[bm * 5 + 2], gy2 = sann[bm * 5 + 3];
      const float gw = fmaxf(gx2 - gx1, 1.0f), gh = fmaxf(gy2 - gy1, 1.0f);
      const float gcx = gx1 + 0.5f * gw, gcy = gy1 + 0.5f * gh;
      float t[4];
      t[0] = (gcx - acx) / aw * 10.0f;   // / 0.1
      t[1] = (gcy - acy) / ah * 10.0f;
      t[2] = logf(gw / aw) * 5.0f;       // / 0.2
      t[3] = logf(gh / ah) * 5.0f;

      const float4* rp = (const float4*)(reg + ((size_t)b * A + a) * 12);
      const float4 r0 = rp[0], r1 = rp[1], r2 = rp[2];
      const float al[12] = {r0.x, r0.y, r0.z, r0.w,
                            r1.x, r1.y, r1.z, r1.w,
                            r2.x, r2.y, r2.z, r2.w};
      const float bins[3] = {0.1667f, 0.5f, 0.8333f};

#pragma unroll
      for (int j = 0; j < 4; ++j) {
        const float p0 = al[3 * j], p1 = al[3 * j + 1], p2 = al[3 * j + 2];
        // predicted delta from mean of the 3-bin Dirichlet
        const float s = fmaxf(p0 + p1 + p2, 1e-6f);
        float c01 = (p0 * bins[0] + p1 * bins[1] + p2 * bins[2]) / s;
        c01 = fminf(fmaxf(c01, 0.0f), 1.0f);
        const float pd = (c01 - 0.5f) * 6.0f;
        const float diff = fabsf(t[j] - pd);
        reg_acc += (diff <= (1.0f / 9.0f)) ? 4.5f * diff * diff
                                           : diff - (0.5f / 9.0f);
        l1_acc += diff;

        // target alphas from GT delta (searchsorted left + interpolation)
        const float tn = fminf(fmaxf(t[j], -3.0f), 3.0f) * (1.0f / 6.0f) + 0.5f;
        const int idx = (int)(tn > bins[0]) + (int)(tn > bins[1]) + (int)(tn > bins[2]);
        const int lo = min(max(idx - 1, 0), 2);
        const int up = min(idx, 2);
        const float lb = bins[lo], ub = bins[up];
        const float wu = (ub > lb) ? (tn - lb) / (ub - lb) : 0.0f;
        const float wl = 1.0f - wu;
        float q[3] = {1e-3f, 1e-3f, 1e-3f};
        q[lo] += 20.0f * wl;
        q[up] += 20.0f * wu;

        // KL(Dir(p) || Dir(q))
        const float pc0 = fminf(fmaxf(p0, 1e-4f), 1e4f);
        const float pc1 = fminf(fmaxf(p1, 1e-4f), 1e4f);
        const float pc2 = fminf(fmaxf(p2, 1e-4f), 1e4f);
        const float psum = fminf(fmaxf(pc0 + pc1 + pc2, 1e-4f), 1e4f);
        const float qsum = fminf(fmaxf(q[0] + q[1] + q[2], 1e-4f), 1e4f);
        const float term1 = lgammaf(psum) - lgammaf(qsum);
        const float term2 = (lgammaf(q[0]) - lgammaf(pc0))
                          + (lgammaf(q[1]) - lgammaf(pc1))
                          + (lgammaf(q[2]) - lgammaf(pc2));
        const float dps = digammaf_(psum);
        const float term3 = (pc0 - q[0]) * (digammaf_(pc0) - dps)
                          + (pc1 - q[1]) * (digammaf_(pc1) - dps)
                          + (pc2 - q[2]) * (digammaf_(pc2) - dps);
        float kl = term1 + term2 + term3;
        if (!(kl == kl)) kl = 0.0f;               // nan_to_num
        kl = fminf(fmaxf(kl, 0.0f), 1e6f);
        kl_acc += kl;
      }
    }
  }

  // Uniform (EXEC all-ones) wave-level WMMA reductions of the 5 partials.
  const float s0 = wave_sum_wmma(cls_acc);
  const float s1 = wave_sum_wmma(reg_acc);
  const float s2 = wave_sum_wmma(kl_acc);
  const float s3 = wave_sum_wmma(l1_acc);
  const float s4 = wave_sum_wmma(npos);

  if ((tid & 31) == 0) {
    float* wb = ws + b * 8;
    atomicAdd(wb + 0, s0);
    atomicAdd(wb + 1, s1);
    atomicAdd(wb + 2, s2);
    atomicAdd(wb + 3, s3);
    atomicAdd(wb + 4, s4);
  }
}

__global__ void dfl_final(const float* __restrict__ ws,
                          const float* __restrict__ ann,
                          float* __restrict__ out,
                          int B, int M) {
  if (threadIdx.x == 0 && blockIdx.x == 0) {
    float o0 = 0.0f, o1 = 0.0f, o2 = 0.0f, o3 = 0.0f;
    for (int b = 0; b < B; ++b) {
      const float np = fmaxf(ws[b * 8 + 4], 1.0f);
      float has = 0.0f;
      for (int m = 0; m < M; ++m)
        if (ann[(size_t)b * M * 5 + m * 5 + 4] != -1.0f) { has = 1.0f; break; }
      o0 += ws[b * 8 + 0] / np;
      const float d4 = np * 4.0f;
      o1 += (ws[b * 8 + 1] / d4) * has;
      o2 += 0.005f * (ws[b * 8 + 2] / d4) * has;   // KL_WEIGHT
      o3 += (ws[b * 8 + 3] / d4) * has;            // L1_WEIGHT = 1
    }
    const float invB = 1.0f / (float)B;
    out[0] = o0 * invB;
    out[1] = o1 * invB;
    out[2] = o2 * invB;
    out[3] = o3 * invB;
  }
}

extern "C" void kernel_launch(void* const* d_in, const int* in_sizes, int n_in,
                              void* d_out, int out_size, void* d_ws, size_t ws_size,
                              hipStream_t stream) {
  const float* cls     = (const float*)d_in[0];  // [B, A, 80]
  const float* reg     = (const float*)d_in[1];  // [B, A, 12]
  const float* anchors = (const float*)d_in[2];  // [1, A, 4]
  const float* ann     = (const float*)d_in[3];  // [B, M, 5]
  float* out = (float*)d_out;                    // 4 floats
  float* ws  = (float*)d_ws;

  const int A = in_sizes[2] / 4;
  const int B = (int)(in_sizes[0] / ((long long)A * NUM_CLASSES));
  const int M = in_sizes[3] / (B * 5);

  hipMemsetAsync(ws, 0, (size_t)B * 8 * sizeof(float), stream);

  dim3 block(256);
  dim3 grid((A + 255) / 256, B);
  const size_t shmem = (size_t)M * 5 * sizeof(float);
  dfl_main<<<grid, block, shmem, stream>>>(cls, reg, anchors, ann, ws, A, M);
  dfl_final<<<1, 32, 0, stream>>>(ws, ann, out, B, M);
}